// HouseholderRotation_7851200217990
// MI455X (gfx1250) — compile-verified
//
#include <hip/hip_runtime.h>

#define DDIM 2048
#define NV 16
#define CPAD 17   // padded row stride for C scratch in LDS (bank-conflict avoidance)

typedef float v2f __attribute__((ext_vector_type(2)));
typedef float v8f __attribute__((ext_vector_type(8)));

// ---------------------------------------------------------------------------
// Kernel 0: build Tneg = -T for the compact WY form  P1..P16 = I - Y T Y^T.
// One block of 256 threads: thread t computes G[t>>4][t&15] = v_i . v_j,
// then thread 0 runs the tiny 16x16 triangular recursion.
// ---------------------------------------------------------------------------
__global__ void __launch_bounds__(256) hh_build_T(const float* __restrict__ vecs,
                                                  float* __restrict__ Tneg_out) {
    __shared__ float G[NV * NV];
    __shared__ float T[NV * NV];
    const int t = threadIdx.x;
    const int i = t >> 4;
    const int j = t & 15;
    const float* vi = vecs + i * DDIM;
    const float* vj = vecs + j * DDIM;
    float s = 0.f;
    for (int d = 0; d < DDIM; d += 4) {
        float4 a = *(const float4*)(vi + d);
        float4 b = *(const float4*)(vj + d);
        s += a.x * b.x + a.y * b.y + a.z * b.z + a.w * b.w;
    }
    G[i * NV + j] = s;
    __syncthreads();
    if (t == 0) {
        float beta[NV];
        for (int k = 0; k < NV; ++k)
            beta[k] = 2.0f / fmaxf(G[k * NV + k], 1e-8f);
        for (int a = 0; a < NV * NV; ++a) T[a] = 0.f;
        T[0] = beta[0];
        for (int k = 1; k < NV; ++k) {
            for (int r = 0; r < k; ++r) {
                float u = 0.f;
                for (int c = r; c < k; ++c)       // T is upper triangular
                    u += T[r * NV + c] * G[c * NV + k];
                T[r * NV + k] = -beta[k] * u;
            }
            T[k * NV + k] = beta[k];
        }
        for (int a = 0; a < NV * NV; ++a) Tneg_out[a] = -T[a];
    }
}

// ---------------------------------------------------------------------------
// Main kernel: one wave32 per 16-row tile of h.
//   Phase 1: C = H_tile x Y            (512 x v_wmma_f32_16x16x4_f32, K=2048)
//   Mid:     W' = C x (-T)             (tiny, via LDS scratch, into A-layout)
//   Phase 2: Out = H_tile + W' x Y^T   (4 WMMAs per 16-col chunk)
// ---------------------------------------------------------------------------
__global__ void __launch_bounds__(256) hh_apply(const float* __restrict__ h,
                                                const float* __restrict__ vecs,
                                                const float* __restrict__ Tneg,
                                                float* __restrict__ out) {
    __shared__ float ldsT[NV * NV];
    __shared__ float ldsC[8 * NV * CPAD];   // per-wave C scratch, padded rows

    const int tid  = threadIdx.x;
    const int wave = tid >> 5;
    const int lane = tid & 31;
    const int m    = lane & 15;   // A/C: row-in-tile. B: column index.
    const int half = lane >> 4;   // 0/1
    const int kOff = half * 2;    // K sub-offset for 16x16x4 f32 fragments

    if (tid < NV * NV) ldsT[tid] = Tneg[tid];
    __syncthreads();

    const int rowBase = (blockIdx.x * 8 + wave) * 16;

    // ---- Phase 1: C[m][j] = sum_d h[rowBase+m][d] * vecs[j][d] ----
    // A fragment: lane -> h row (rowBase+m), K = d0+kOff, d0+kOff+1
    // B fragment: lane -> vector j=m,       K = d0+kOff, d0+kOff+1
    const float* hA = h + (size_t)(rowBase + m) * DDIM + kOff;
    const float* vB = vecs + (size_t)m * DDIM + kOff;

    v8f acc = {};
    #pragma unroll 8
    for (int d0 = 0; d0 < DDIM; d0 += 4) {
        v2f a = *(const v2f*)(hA + d0);
        v2f b = *(const v2f*)(vB + d0);
        acc = __builtin_amdgcn_wmma_f32_16x16x4_f32(false, a, false, b,
                                                    (short)0, acc, false, false);
    }

    // ---- Spill C to LDS (C layout: VGPR r -> row r+8*half, col m) ----
    float* Cs = &ldsC[wave * NV * CPAD];
    #pragma unroll
    for (int r = 0; r < 8; ++r)
        Cs[(r + 8 * half) * CPAD + m] = acc[r];
    __syncthreads();

    // ---- W' = C x Tneg, produced directly in A-fragment layout ----
    // Lane needs W'[m][k0+kOff+{0,1}] for k0 in {0,4,8,12}.
    v2f wA[4];
    #pragma unroll
    for (int t4 = 0; t4 < 4; ++t4) {
        const int j0 = 4 * t4 + kOff;
        float s0 = 0.f, s1 = 0.f;
        #pragma unroll
        for (int i2 = 0; i2 < NV; ++i2) {
            const float c = Cs[m * CPAD + i2];
            s0 += c * ldsT[i2 * NV + j0];
            s1 += c * ldsT[i2 * NV + j0 + 1];
        }
        wA[t4].x = s0;
        wA[t4].y = s1;
    }

    // ---- Phase 2: out[m][n] = h[m][n] + sum_j W'[m][j] * vecs[j][n] ----
    const size_t base = (size_t)rowBase * DDIM;
    for (int n0 = 0; n0 < DDIM; n0 += 16) {
        const int col = n0 + m;
        v8f cacc;
        #pragma unroll
        for (int r = 0; r < 8; ++r)                         // h tile in C layout
            cacc[r] = h[base + (size_t)(r + 8 * half) * DDIM + col];
        #pragma unroll
        for (int t4 = 0; t4 < 4; ++t4) {                    // B[k][n]=vecs[4t+k][n0+n]
            v2f b;
            b.x = vecs[(size_t)(4 * t4 + kOff)     * DDIM + col];
            b.y = vecs[(size_t)(4 * t4 + kOff + 1) * DDIM + col];
            cacc = __builtin_amdgcn_wmma_f32_16x16x4_f32(false, wA[t4], false, b,
                                                         (short)0, cacc, false, false);
        }
        #pragma unroll
        for (int r = 0; r < 8; ++r)
            out[base + (size_t)(r + 8 * half) * DDIM + col] = cacc[r];
    }
}

// ---------------------------------------------------------------------------
extern "C" void kernel_launch(void* const* d_in, const int* in_sizes, int n_in,
                              void* d_out, int out_size, void* d_ws, size_t ws_size,
                              hipStream_t stream) {
    const float* h    = (const float*)d_in[0];   // [4,4096,2048] f32
    const float* vecs = (const float*)d_in[1];   // [16,2048] f32
    float* out  = (float*)d_out;
    float* Tneg = (float*)d_ws;                  // 16x16 f32 scratch

    const int rows  = in_sizes[0] / DDIM;        // 16384
    const int tiles = rows / 16;                 // 1024 wave-tiles
    const int blocks = tiles / 8;                // 8 waves (256 thr) per block

    hh_build_T<<<1, 256, 0, stream>>>(vecs, Tneg);
    hh_apply<<<blocks, 256, 0, stream>>>(h, vecs, Tneg, out);
}